// WaveletOptimizedStyleTransfer_57294863729405
// MI455X (gfx1250) — compile-verified
//
#include <hip/hip_runtime.h>
#include <cstdint>
#include <cstddef>

// Problem geometry (fixed by the reference)
#define NROW_ELEMS 65536      // H*W
#define NUM_ROWS   256        // B*C
#define NTOT       16777216   // elements per tensor (B*C*H*W)

#define WG    1024            // sort/search workgroup (32 wave32 waves)
#define SEG   64              // elements per thread in the sort (NROW_ELEMS / WG)
#define RADIX 16

// cos() of the 4 possible blended phases
#define COS_02PI  0.80901699437494745f   // cos(0.2*pi)  (approx band, alpha=0.8)
#define COS_04PI  0.30901699437494745f   // cos(0.4*pi)  (detail bands, alpha=0.4)

// ---- gfx1250 async load-to-LDS availability (device pass only) ----
#if defined(__HIP_DEVICE_COMPILE__)
#if __has_builtin(__builtin_amdgcn_global_load_async_to_lds_b128)
#define HAVE_ASYNC_LDS 1
#endif
#endif

// Builtin signature (verified via compiler diagnostic in a previous round):
//   void __builtin_amdgcn_global_load_async_to_lds_b128(
//       int __attribute__((vector_size(16))) AS1* gsrc,
//       int __attribute__((vector_size(16))) AS3* ldst,
//       imm int offset, imm int cpol)
typedef int v4i_t __attribute__((vector_size(16)));
typedef __attribute__((address_space(1))) v4i_t gv4i_t;   // global
typedef __attribute__((address_space(3))) v4i_t lv4i_t;   // LDS

// true clang vector type so __builtin_nontemporal_* accepts it
typedef float fvec4 __attribute__((ext_vector_type(4)));

__device__ __forceinline__ unsigned absbits(float x) {
  // |x| as sortable key: nonnegative float bit pattern is monotone in value
  return __float_as_uint(x) & 0x7fffffffu;
}

// ---------------------------------------------------------------------------
// K0: build sort keys.  s-keys (|content_approx|) -> bufA, t-keys -> bufC
// Keys are re-read immediately by the sort: keep default (RT) caching.
// ---------------------------------------------------------------------------
__global__ void k_build_keys(const float* __restrict__ ca,
                             const float* __restrict__ sa,
                             unsigned* __restrict__ skeys,
                             unsigned* __restrict__ tkeys) {
  const int idx = (blockIdx.x * blockDim.x + threadIdx.x) * 4;
  if (idx < NTOT) {
    float4 c = *(const float4*)(ca + idx);
    float4 s = *(const float4*)(sa + idx);
    uint4 sk, tk;
    sk.x = absbits(c.x); sk.y = absbits(c.y); sk.z = absbits(c.z); sk.w = absbits(c.w);
    tk.x = absbits(s.x); tk.y = absbits(s.y); tk.z = absbits(s.z); tk.w = absbits(s.w);
    *(uint4*)(skeys + idx) = sk;
    *(uint4*)(tkeys + idx) = tk;
  }
}

// ---------------------------------------------------------------------------
// K1: per-row LSD radix-16 sort (8 passes), one workgroup per row.
// Stable: thread t owns contiguous elements [t*64, t*64+64); cross-thread
// per-digit exclusive prefix via LDS + wave32 shfl scans.
// Even pass count => result ends back in buf0.
// Dynamic LDS: (RADIX*WG + 2*RADIX) u32  (~64.1 KB)
// ---------------------------------------------------------------------------
__global__ __launch_bounds__(WG) void k_radix_sort_rows(unsigned* __restrict__ buf0,
                                                        unsigned* __restrict__ buf1) {
  extern __shared__ unsigned smem[];
  unsigned (*cnt)[WG] = (unsigned (*)[WG])smem;          // [RADIX][WG]
  unsigned* digtot  = smem + RADIX * WG;                 // [RADIX]
  unsigned* digbase = digtot + RADIX;                    // [RADIX]

  const int tid  = threadIdx.x;
  const int lane = tid & 31;
  const int wave = tid >> 5;
  unsigned* A  = buf0 + (size_t)blockIdx.x * NROW_ELEMS;
  unsigned* Bp = buf1 + (size_t)blockIdx.x * NROW_ELEMS;

  for (int pass = 0; pass < 8; ++pass) {
    unsigned* src = (pass & 1) ? Bp : A;
    unsigned* dst = (pass & 1) ? A : Bp;
    const int shift = pass * 4;

#pragma unroll
    for (int d = 0; d < RADIX; ++d) cnt[d][tid] = 0;
    __syncthreads();

    // Phase A: per-thread digit histogram over its contiguous segment
    const uint4* s4 = (const uint4*)(src + tid * SEG);
#pragma unroll 4
    for (int k = 0; k < SEG / 4; ++k) {
      uint4 v = s4[k];
      cnt[(v.x >> shift) & (RADIX - 1)][tid] += 1u;
      cnt[(v.y >> shift) & (RADIX - 1)][tid] += 1u;
      cnt[(v.z >> shift) & (RADIX - 1)][tid] += 1u;
      cnt[(v.w >> shift) & (RADIX - 1)][tid] += 1u;
    }
    __syncthreads();

    // Phase B: exclusive scan over 1024 threads per digit (1 wave per digit)
    if (wave < RADIX) {
      const int d = wave;
      const int base = lane * 32;
      unsigned run = 0;
      for (int k = 0; k < 32; ++k) {        // serial exclusive scan of 32 entries
        unsigned v = cnt[d][base + k];
        cnt[d][base + k] = run;
        run += v;
      }
      unsigned incl = run;                  // wave32 inclusive scan of lane sums
#pragma unroll
      for (int off = 1; off < 32; off <<= 1) {
        unsigned o = __shfl_up(incl, off, 32);
        if (lane >= off) incl += o;
      }
      const unsigned excl = incl - run;
      for (int k = 0; k < 32; ++k) cnt[d][base + k] += excl;
      if (lane == 31) digtot[d] = incl;
    }
    __syncthreads();

    if (tid == 0) {                          // exclusive scan of 16 digit totals
      unsigned b = 0;
      for (int d = 0; d < RADIX; ++d) { digbase[d] = b; b += digtot[d]; }
    }
    __syncthreads();

    // Phase C: stable scatter (thread-major index order == global index order)
    for (int k = 0; k < SEG / 4; ++k) {
      uint4 v = s4[k];
      unsigned e, d, p;
      e = v.x; d = (e >> shift) & (RADIX - 1); p = cnt[d][tid]; cnt[d][tid] = p + 1u; dst[digbase[d] + p] = e;
      e = v.y; d = (e >> shift) & (RADIX - 1); p = cnt[d][tid]; cnt[d][tid] = p + 1u; dst[digbase[d] + p] = e;
      e = v.z; d = (e >> shift) & (RADIX - 1); p = cnt[d][tid]; cnt[d][tid] = p + 1u; dst[digbase[d] + p] = e;
      e = v.w; d = (e >> shift) & (RADIX - 1); p = cnt[d][tid]; cnt[d][tid] = p + 1u; dst[digbase[d] + p] = e;
    }
    __syncthreads();
  }
}

// ---------------------------------------------------------------------------
// K2: histogram match + phase blend for the approx band.
// Whole sorted content row (256 KB) lives in LDS (320 KB/WGP on CDNA5),
// staged via gfx1250 async load-to-LDS.
// out[i] = t_sorted[lower_bound(s_sorted, key(c[i]))] * cos(blended_phase)
// Dynamic LDS: NROW_ELEMS u32 (256 KB)
// ---------------------------------------------------------------------------
__global__ __launch_bounds__(WG) void k_hist_match(const float* __restrict__ ca,
                                                   const float* __restrict__ sa,
                                                   const unsigned* __restrict__ ssorted,
                                                   const unsigned* __restrict__ tsorted,
                                                   float* __restrict__ outA) {
  extern __shared__ unsigned lds[];   // NROW_ELEMS sorted content keys
  const int tid = threadIdx.x;
  const size_t rowoff = (size_t)blockIdx.x * NROW_ELEMS;
  const unsigned* Sg = ssorted + rowoff;
  const unsigned* Tg = tsorted + rowoff;

#if HAVE_ASYNC_LDS
#pragma unroll
  for (int k = 0; k < NROW_ELEMS / (WG * 4); ++k) {
    const int idx = (k * WG + tid) * 4;
    unsigned long long ga = (unsigned long long)(uintptr_t)(const void*)(Sg + idx);
    unsigned           la = (unsigned)(uintptr_t)(void*)(lds + idx);
    __builtin_amdgcn_global_load_async_to_lds_b128((gv4i_t*)ga, (lv4i_t*)la, 0, 0);
  }
#if __has_builtin(__builtin_amdgcn_s_wait_asynccnt)
  __builtin_amdgcn_s_wait_asynccnt(0);
#else
  asm volatile("s_wait_asynccnt 0" ::: "memory");
#endif
  __syncthreads();
#else
  for (int k = 0; k < NROW_ELEMS / (WG * 4); ++k) {
    const int idx = (k * WG + tid) * 4;
    *(uint4*)(lds + idx) = *(const uint4*)(Sg + idx);
  }
  __syncthreads();
#endif

  for (int k = 0; k < NROW_ELEMS / (WG * 4); ++k) {
    const int i = (k * WG + tid) * 4;
    float4 c = *(const float4*)(ca + rowoff + i);
    float4 s = *(const float4*)(sa + rowoff + i);
    float cc[4] = {c.x, c.y, c.z, c.w};
    float ss[4] = {s.x, s.y, s.z, s.w};
    unsigned pos[4];
#pragma unroll
    for (int e = 0; e < 4; ++e) {           // 4 independent searches for ILP
      const unsigned key = absbits(cc[e]);
      unsigned p = 0;
#pragma unroll
      for (int half = NROW_ELEMS >> 1; half > 0; half >>= 1)
        if (lds[p + half - 1] < key) p += half;   // key present => p <= 65535
      pos[e] = p;
    }
    float ov[4];
#pragma unroll
    for (int e = 0; e < 4; ++e) {
      const float mag = __uint_as_float(Tg[pos[e]]);   // matched magnitude
      const float f = (cc[e] < 0.0f) ? ((ss[e] < 0.0f) ? -1.0f : -COS_02PI)
                                     : ((ss[e] < 0.0f) ?  COS_02PI : 1.0f);
      ov[e] = mag * f;
    }
    float4 o; o.x = ov[0]; o.y = ov[1]; o.z = ov[2]; o.w = ov[3];
    *(float4*)(outA + rowoff + i) = o;
  }
}

// ---------------------------------------------------------------------------
// K3: detail bands — pure single-touch streaming; use nontemporal (TH=NT)
// loads/stores so 603 MB of stream traffic doesn't thrash caches.
// ---------------------------------------------------------------------------
__device__ __forceinline__ float detail1(float c, float s) {
  const float mag = 0.4f * fabsf(c) + 0.6f * fabsf(s);
  const float f = (c < 0.0f) ? ((s < 0.0f) ? -1.0f : COS_04PI)
                             : ((s < 0.0f) ? -COS_04PI : 1.0f);
  return mag * f;
}

__global__ void k_detail(const float* __restrict__ c,
                         const float* __restrict__ s,
                         float* __restrict__ o) {
  const int idx = (blockIdx.x * blockDim.x + threadIdx.x) * 4;
  if (idx < NTOT) {
    __builtin_prefetch(c + idx + 8192, 0, 1);   // global_prefetch_b8 (CDNA5)
    __builtin_prefetch(s + idx + 8192, 0, 1);
    fvec4 cv = __builtin_nontemporal_load((const fvec4*)(c + idx));
    fvec4 sv = __builtin_nontemporal_load((const fvec4*)(s + idx));
    fvec4 ov;
    ov.x = detail1(cv.x, sv.x);
    ov.y = detail1(cv.y, sv.y);
    ov.z = detail1(cv.z, sv.z);
    ov.w = detail1(cv.w, sv.w);
    __builtin_nontemporal_store(ov, (fvec4*)(o + idx));
  }
}

// ---------------------------------------------------------------------------
// Host launcher.
// Scratch: the h/v/d regions of d_out are used as the three 64 MB sort
// buffers (ping/pong + t keys), then fully overwritten by the detail kernels.
// Every buffer is fully rewritten each call -> deterministic, graph-safe.
// ---------------------------------------------------------------------------
extern "C" void kernel_launch(void* const* d_in, const int* in_sizes, int n_in,
                              void* d_out, int out_size, void* d_ws, size_t ws_size,
                              hipStream_t stream) {
  (void)in_sizes; (void)n_in; (void)out_size; (void)d_ws; (void)ws_size;

  const float* ca = (const float*)d_in[0];   // content_approx
  const float* ch = (const float*)d_in[1];   // content_h
  const float* cv = (const float*)d_in[2];   // content_v
  const float* cd = (const float*)d_in[3];   // content_d
  const float* sa = (const float*)d_in[4];   // style_approx
  const float* sh = (const float*)d_in[5];   // style_h
  const float* sv = (const float*)d_in[6];   // style_v
  const float* sd = (const float*)d_in[7];   // style_d

  float* out  = (float*)d_out;
  float* outA = out;                         // stylized_approx
  float* outH = out + (size_t)NTOT;          // stylized_h
  float* outV = out + (size_t)2 * NTOT;      // stylized_v
  float* outD = out + (size_t)3 * NTOT;      // stylized_d

  unsigned* bufA = (unsigned*)outH;          // content keys -> s_sorted
  unsigned* bufB = (unsigned*)outV;          // ping/pong
  unsigned* bufC = (unsigned*)outD;          // style keys   -> t_sorted

  const int ewBlocks = NTOT / 4 / 256;                       // 16384
  const int sortLds   = (int)((RADIX * WG + 2 * RADIX) * sizeof(unsigned)); // 65664 B
  const int searchLds = (int)(NROW_ELEMS * sizeof(unsigned));               // 256 KB

  // Raise the dynamic-LDS cap (CDNA5 WGP has 320 KB; runtime default may be
  // lower). Deterministic + capture-safe: not a stream operation.
  (void)hipFuncSetAttribute((const void*)k_radix_sort_rows,
                            hipFuncAttributeMaxDynamicSharedMemorySize, sortLds);
  (void)hipFuncSetAttribute((const void*)k_hist_match,
                            hipFuncAttributeMaxDynamicSharedMemorySize, searchLds);

  k_build_keys<<<ewBlocks, 256, 0, stream>>>(ca, sa, bufA, bufC);
  k_radix_sort_rows<<<NUM_ROWS, WG, sortLds, stream>>>(bufA, bufB);  // sort |content|
  k_radix_sort_rows<<<NUM_ROWS, WG, sortLds, stream>>>(bufC, bufB);  // sort |style|
  k_hist_match<<<NUM_ROWS, WG, searchLds, stream>>>(ca, sa, bufA, bufC, outA);
  k_detail<<<ewBlocks, 256, 0, stream>>>(ch, sh, outH);
  k_detail<<<ewBlocks, 256, 0, stream>>>(cv, sv, outV);
  k_detail<<<ewBlocks, 256, 0, stream>>>(cd, sd, outD);
}